// MyGraphConvModel_85229331022471
// MI455X (gfx1250) — compile-verified
//
#include <hip/hip_runtime.h>
#include <hip/hip_bf16.h>

// ---------------------------------------------------------------------------
// GraphConv pipeline for MI455X (gfx1250, wave32, WMMA).
// 4 fused gather+GEMM+SELU conv layers, segment sum/max pooling, dense+softmax.
// Conv kernel is templated on <FI, FO, DEG> so the V_WMMA_F32_16X16X4_F32
// K-loop is fully static and branch-free; A/B panels staged in LDS, with the
// B panel pair-interleaved so every fragment is one aligned ds_load_b64.
// ---------------------------------------------------------------------------

#define MAX_DEG   10
#define N_ATOMS   600000
#define BATCH     24000
#define N_TASKS   12
#define TILE_M    16
#define WAVES     4

typedef __attribute__((ext_vector_type(2))) float v2f;
typedef __attribute__((ext_vector_type(8))) float v8f;

static const int H_DEG_COUNTS[MAX_DEG + 1] =
    {10000, 130000, 170000, 160000, 100000, 15000, 5000, 2500, 2500, 2500, 2500};
static const int H_DEG_STARTS[MAX_DEG + 1] =
    {0, 10000, 140000, 310000, 470000, 570000, 585000, 590000, 592500, 595000, 597500};

__device__ __forceinline__ float selu_f(float x) {
    const float alpha = 1.6732632423543772f;
    const float scale = 1.0507009873554805f;
    return x > 0.f ? scale * x : scale * alpha * (expf(x) - 1.f);
}

// Fused: rel = sum(xin[adj], axis=1); out = selu(rel@Wa + ba + self@Wb + bb)
// DEG==0: out = selu(self@Wa + ba).
// One wave per 16-atom tile; 4 waves (128 threads) per block.
template<int FI, int FO, int DEG>
__global__ __launch_bounds__(128) void gconv_kernel(
    const float* __restrict__ xin, const int* __restrict__ adj,
    const float* __restrict__ Wa, const float* __restrict__ Wb,
    const float* __restrict__ ba, const float* __restrict__ bb,
    float* __restrict__ xout, int start, int count)
{
    constexpr int KLOG = (DEG > 0) ? 2 * FI : FI;
    constexpr int KPAD = (KLOG + 3) & ~3;
    constexpr int NT   = (FO + 15) / 16;      // <= 3
    constexpr int NPAD = NT * 16;

    __shared__ float A_lds[WAVES][TILE_M * KPAD];
    __shared__ float B_lds[KPAD * NPAD];      // pair-interleaved: ((k/2)*NPAD+n)*2+(k&1)

    const int lane = threadIdx.x & 31;
    const int wave = threadIdx.x >> 5;
    const int base = (blockIdx.x * WAVES + wave) * TILE_M;
    float* Aw = &A_lds[wave][0];

    if constexpr (DEG > 0) {
        if (base < count)
            __builtin_prefetch(adj + (size_t)base * DEG, 0, 3);  // global_prefetch_b8
    }

    // ---- stage B = [Wa; Wb] (zero-padded, pair-interleaved) into LDS -------
    for (int idx = threadIdx.x; idx < KPAD * NPAD; idx += 128) {
        const int k = idx / NPAD;             // constexpr divisor
        const int n = idx - k * NPAD;
        float v = 0.f;
        if (n < FO) {
            if (k < FI)                       v = Wa[k * FO + n];
            else if (DEG > 0 && k < 2 * FI)   v = Wb[(k - FI) * FO + n];
        }
        B_lds[((k >> 1) * NPAD + n) * 2 + (k & 1)] = v;
    }

    // ---- gather phase: build A = [rel | self] (or [self]) in LDS -----------
    for (int row = 0; row < TILE_M; ++row) {
        const int a = base + row;
        const bool valid = (a < count);                 // wave-uniform
        if constexpr (DEG > 0) {
            const int* arow = adj + (size_t)a * DEG;
            for (int k = lane; k < FI; k += 32) {       // rel part
                float s = 0.f;
                if (valid) {
#pragma unroll
                    for (int j = 0; j < DEG; ++j)
                        s += xin[(size_t)arow[j] * FI + k];
                }
                Aw[row * KPAD + k] = s;
            }
            for (int k = FI + lane; k < KPAD; k += 32) {  // self part + pad
                float v = (valid && k < KLOG)
                        ? xin[(size_t)(start + a) * FI + (k - FI)] : 0.f;
                Aw[row * KPAD + k] = v;
            }
        } else {
            for (int k = lane; k < KPAD; k += 32) {       // self only + pad
                float v = (valid && k < FI)
                        ? xin[(size_t)(start + a) * FI + k] : 0.f;
                Aw[row * KPAD + k] = v;
            }
        }
    }
    __syncthreads();

    // ---- WMMA phase: fully static, branch-free K-loop ----------------------
    const int halfsel = lane >> 4;            // 0: lanes 0-15, 1: lanes 16-31
    const int l15     = lane & 15;

    v8f acc[NT] = {};

    for (int k0 = 0; k0 < KPAD; k0 += 4) {
        // A frag (f32 16x4): lanes 0-15 K=k0,k0+1; lanes 16-31 K=k0+2,k0+3
        const int kb = k0 + (halfsel << 1);   // even -> 8B aligned
        const v2f afrag = *(const v2f*)&Aw[l15 * KPAD + kb];
#pragma unroll
        for (int nt = 0; nt < NT; ++nt) {
            const int n = nt * 16 + l15;
            const v2f bfrag = *(const v2f*)&B_lds[((kb >> 1) * NPAD + n) * 2];
            acc[nt] = __builtin_amdgcn_wmma_f32_16x16x4_f32(
                false, afrag, false, bfrag, (short)0, acc[nt], false, false);
        }
    }

    // ---- epilogue: bias + SELU + store --------------------------------------
#pragma unroll
    for (int nt = 0; nt < NT; ++nt) {
        const int n = nt * 16 + l15;
        const bool ncol = (n < FO);
        const float biasv = ncol ? (ba[n] + (DEG > 0 ? bb[n] : 0.f)) : 0.f;
#pragma unroll
        for (int r = 0; r < 8; ++r) {
            const int m = r + (halfsel << 3); // VGPR r: lanes 0-15 M=r, 16-31 M=r+8
            const int a = base + m;
            if (a < count && ncol) {
                xout[(size_t)(start + a) * FO + n] = selu_f(acc[nt][r] + biasv);
            }
        }
    }
}

// ---------------------------------------------------------------------------
__global__ void init_pool_kernel(float* __restrict__ sums, float* __restrict__ maxs, int n) {
    int i = blockIdx.x * blockDim.x + threadIdx.x;
    if (i < n) { sums[i] = 0.f; maxs[i] = -INFINITY; }
}

__device__ __forceinline__ void atomicMaxFloat(float* addr, float val) {
    if (!(val == val)) return;           // skip NaN
    if (val >= 0.f)  atomicMax((int*)addr, __float_as_int(val));
    else             atomicMin((unsigned int*)addr, __float_as_uint(val));
}

__global__ void pool_kernel(const float* __restrict__ x, const int* __restrict__ membership,
                            float* __restrict__ sums, float* __restrict__ maxs,
                            int natoms, int feat) {
    int i = blockIdx.x * blockDim.x + threadIdx.x;
    if (i >= natoms * feat) return;
    int atom = i / feat;
    int f    = i - atom * feat;
    float v  = x[i];
    int m    = membership[atom];
    atomicAdd(&sums[(size_t)m * feat + f], v);
    atomicMaxFloat(&maxs[(size_t)m * feat + f], v);
}

// mol = tanh([sums, maxs]); logits = mol @ Wd + bd; softmax over task pairs
__global__ __launch_bounds__(128) void dense_softmax_kernel(
    const float* __restrict__ sums, const float* __restrict__ maxs,
    const float* __restrict__ Wd, const float* __restrict__ bd,
    float* __restrict__ out) {
    int b = blockIdx.x * blockDim.x + threadIdx.x;
    if (b >= BATCH) return;
    float mol[72];
#pragma unroll
    for (int f = 0; f < 36; ++f) mol[f]      = tanhf(sums[(size_t)b * 36 + f]);
#pragma unroll
    for (int f = 0; f < 36; ++f) mol[36 + f] = tanhf(maxs[(size_t)b * 36 + f]);
#pragma unroll
    for (int t = 0; t < N_TASKS; ++t) {
        float l0 = bd[2 * t], l1 = bd[2 * t + 1];
        for (int f = 0; f < 72; ++f) {
            l0 = fmaf(mol[f], Wd[(size_t)f * 24 + 2 * t], l0);
            l1 = fmaf(mol[f], Wd[(size_t)f * 24 + 2 * t + 1], l1);
        }
        float mx = fmaxf(l0, l1);
        float e0 = expf(l0 - mx), e1 = expf(l1 - mx);
        float inv = 1.f / (e0 + e1);
        out[(size_t)b * 24 + 2 * t]     = e0 * inv;
        out[(size_t)b * 24 + 2 * t + 1] = e1 * inv;
    }
}

// ---------------------------------------------------------------------------
static inline int gconv_grid(int count) {
    const int tiles = (count + TILE_M - 1) / TILE_M;
    return (tiles + WAVES - 1) / WAVES;
}

template<int FI, int FO>
static void launch_layer(const float* xin, const int* const* adjp,
                         const float* Wl, const float* Bl, float* xout,
                         hipStream_t stream)
{
    // degree 0: self-only (uses W[20], b[20])
    gconv_kernel<FI, FO, 0><<<gconv_grid(H_DEG_COUNTS[0]), 128, 0, stream>>>(
        xin, nullptr,
        Wl + (size_t)(2 * MAX_DEG) * FI * FO, nullptr,
        Bl + (size_t)(2 * MAX_DEG) * FO, nullptr,
        xout, H_DEG_STARTS[0], H_DEG_COUNTS[0]);

#define LAUNCH_DEG(D)                                                          \
    gconv_kernel<FI, FO, D><<<gconv_grid(H_DEG_COUNTS[D]), 128, 0, stream>>>(  \
        xin, adjp[D - 1],                                                      \
        Wl + (size_t)(2 * (D - 1)) * FI * FO,                                  \
        Wl + (size_t)(2 * (D - 1) + 1) * FI * FO,                              \
        Bl + (size_t)(2 * (D - 1)) * FO,                                       \
        Bl + (size_t)(2 * (D - 1) + 1) * FO,                                   \
        xout, H_DEG_STARTS[D], H_DEG_COUNTS[D])

    LAUNCH_DEG(1); LAUNCH_DEG(2); LAUNCH_DEG(3); LAUNCH_DEG(4); LAUNCH_DEG(5);
    LAUNCH_DEG(6); LAUNCH_DEG(7); LAUNCH_DEG(8); LAUNCH_DEG(9); LAUNCH_DEG(10);
#undef LAUNCH_DEG
}

extern "C" void kernel_launch(void* const* d_in, const int* in_sizes, int n_in,
                              void* d_out, int out_size, void* d_ws, size_t ws_size,
                              hipStream_t stream) {
    static const int CH[5] = {75, 15, 20, 27, 36};

    const float* atom_features = (const float*)d_in[0];
    const int*   membership    = (const int*)d_in[2];
    const int*   adj[MAX_DEG];
    for (int d = 0; d < MAX_DEG; ++d) adj[d] = (const int*)d_in[3 + d];
    const float* W[4]; const float* B[4];
    for (int l = 0; l < 4; ++l) { W[l] = (const float*)d_in[13 + 2 * l]; B[l] = (const float*)d_in[14 + 2 * l]; }
    const float* Wd = (const float*)d_in[21];
    const float* bd = (const float*)d_in[22];
    float* out = (float*)d_out;

    // workspace layout (bytes)
    char* ws = (char*)d_ws;
    size_t off = 0;
    float* xbuf[5];
    xbuf[0] = nullptr; // input
    for (int l = 1; l <= 4; ++l) {
        xbuf[l] = (float*)(ws + off);
        off += (size_t)N_ATOMS * CH[l] * sizeof(float);
    }
    float* sums = (float*)(ws + off); off += (size_t)BATCH * 36 * sizeof(float);
    float* maxs = (float*)(ws + off); off += (size_t)BATCH * 36 * sizeof(float);
    (void)ws_size; (void)in_sizes; (void)n_in; (void)out_size;

    // ---- 4 graph conv layers, fused per (layer, degree) ----
    launch_layer<75, 15>(atom_features, adj, W[0], B[0], xbuf[1], stream);
    launch_layer<15, 20>(xbuf[1],       adj, W[1], B[1], xbuf[2], stream);
    launch_layer<20, 27>(xbuf[2],       adj, W[2], B[2], xbuf[3], stream);
    launch_layer<27, 36>(xbuf[3],       adj, W[3], B[3], xbuf[4], stream);

    // ---- pooling (segment sum / max) ----
    {
        const int n = BATCH * 36;
        init_pool_kernel<<<(n + 255) / 256, 256, 0, stream>>>(sums, maxs, n);
        const int total = N_ATOMS * 36;
        pool_kernel<<<(total + 255) / 256, 256, 0, stream>>>(
            xbuf[4], membership, sums, maxs, N_ATOMS, 36);
    }

    // ---- dense + softmax ----
    dense_softmax_kernel<<<(BATCH + 127) / 128, 128, 0, stream>>>(sums, maxs, Wd, bd, out);
}